// SimpleHierachicalAttention_61220463837948
// MI455X (gfx1250) — compile-verified
//
#include <hip/hip_runtime.h>
#include <hip/hip_bf16.h>
#include <math.h>

// Problem constants (from reference): B=32, S=2048, H=1024, UNITS=128
#define BB 32
#define SS 2048
#define HH 1024
#define UU 128
#define NSC 16   // s-chunks in context pass

typedef __attribute__((ext_vector_type(2))) float v2f;
typedef __attribute__((ext_vector_type(8))) float v8f;

// ---------------------------------------------------------------------------
// Kernel 1: v[b,h] = sum_k w_score[h,k] * h_t[b,k], h_t[b,k] = hs[b, S-1, k]
// GEMM view: C(32 x 1024) = A(32 x 1024) * B'(1024 x 1024), B'[k][h]=w_score[h][k]
// WMMA f32 16x16x4, one wave per 16x16 tile. Tiles: 2 (M) x 64 (N) = 128.
// ---------------------------------------------------------------------------
__global__ void kv_wmma(const float* __restrict__ hs,
                        const float* __restrict__ w_score,
                        float* __restrict__ v_out) {
    const int lane = threadIdx.x & 31;
    const int wave = threadIdx.x >> 5;            // 0..7
    const int tile = blockIdx.x * 8 + wave;       // 0..127
    const int mt = tile >> 6;                     // 0..1
    const int nt = tile & 63;                     // 0..63

    const int mrow = mt * 16 + (lane & 15);       // batch row for A frag
    const int hcol = nt * 16 + (lane & 15);       // output column (h)
    const int koff = (lane >> 4) * 2;             // lanes 16-31 hold K=2,3

    const float* arow = hs + ((size_t)mrow * SS + (SS - 1)) * HH; // h_t[mrow][*]
    const float* brow = w_score + (size_t)hcol * HH;              // w_score[hcol][*]

    v8f acc = {};
    for (int k = 0; k < HH; k += 4) {
        v2f a = *(const v2f*)(arow + k + koff);   // {A[m][k+koff], A[m][k+koff+1]}
        v2f b = *(const v2f*)(brow + k + koff);   // {B'[k+koff][n], B'[k+koff+1][n]}
        acc = __builtin_amdgcn_wmma_f32_16x16x4_f32(
            false, a, false, b, (short)0, acc, false, false);
    }

    // C/D layout: VGPR i -> lanes 0-15: M=i, lanes 16-31: M=i+8; N = lane&15
    const int ncol  = nt * 16 + (lane & 15);
    const int mbase = mt * 16 + ((lane >> 4) ? 8 : 0);
#pragma unroll
    for (int i = 0; i < 8; ++i)
        v_out[(size_t)(mbase + i) * HH + ncol] = acc[i];
}

// ---------------------------------------------------------------------------
// Kernel 2: score[b,s] = dot(hs[b,s,:], v[b,:]).  Streaming pass #1 (256 MB).
// Block = 256 threads (8 waves); wave handles 8 s-rows; v[b] staged in LDS.
// grid = (S/64, B). Ascending batch order: warms L2 for kpctx's reverse walk.
// ---------------------------------------------------------------------------
__global__ void kscore(const float* __restrict__ hs,
                       const float* __restrict__ v,
                       float* __restrict__ score) {
    __shared__ float v_lds[HH];
    const int b = blockIdx.y;
    const int t = threadIdx.x;
    ((float4*)v_lds)[t] = ((const float4*)(v + (size_t)b * HH))[t]; // 256*4 = 1024
    __syncthreads();

    const int lane = t & 31;
    const int wave = t >> 5;
    const int s0 = blockIdx.x * 64 + wave * 8;
    const float4* vl = (const float4*)v_lds;

    for (int j = 0; j < 8; ++j) {
        const int s = s0 + j;
        const float4* row = (const float4*)(hs + ((size_t)b * SS + s) * HH);
        float acc = 0.f;
#pragma unroll
        for (int i = 0; i < 8; ++i) {
            float4 a = row[lane + i * 32];   // coalesced 512B per wave step
            float4 w = vl[lane + i * 32];
            acc += a.x * w.x + a.y * w.y + a.z * w.z + a.w * w.w;
        }
#pragma unroll
        for (int off = 16; off > 0; off >>= 1)
            acc += __shfl_xor(acc, off, 32);
        if (lane == 0) score[(size_t)b * SS + s] = acc;
    }
}

// ---------------------------------------------------------------------------
// Kernel 3: softmax over s (2048) per batch. In-place (score -> attn).
// ---------------------------------------------------------------------------
__global__ void ksoftmax(float* __restrict__ buf) {
    __shared__ float red[256];
    const int b = blockIdx.x;
    const int t = threadIdx.x;
    float vals[8];
    float m = -INFINITY;
#pragma unroll
    for (int i = 0; i < 8; ++i) {
        vals[i] = buf[(size_t)b * SS + t + i * 256];
        m = fmaxf(m, vals[i]);
    }
    red[t] = m; __syncthreads();
    for (int off = 128; off > 0; off >>= 1) {
        if (t < off) red[t] = fmaxf(red[t], red[t + off]);
        __syncthreads();
    }
    m = red[0]; __syncthreads();
    float sum = 0.f;
#pragma unroll
    for (int i = 0; i < 8; ++i) { vals[i] = expf(vals[i] - m); sum += vals[i]; }
    red[t] = sum; __syncthreads();
    for (int off = 128; off > 0; off >>= 1) {
        if (t < off) red[t] += red[t + off];
        __syncthreads();
    }
    const float inv = 1.f / red[0];
#pragma unroll
    for (int i = 0; i < 8; ++i)
        buf[(size_t)b * SS + t + i * 256] = vals[i] * inv;
}

// ---------------------------------------------------------------------------
// Kernel 4: partial context. Streaming pass #2.
// grid = (NSC s-chunks, B), block = 256 threads; thread owns 4 consecutive h
// (float4 -> global_load_b128), block covers all 1024 h; 128 s per chunk.
// Batch order REVERSED vs kscore so the ~192MB L2 still holds the tail of hs
// -> only ~64MB of this pass misses to HBM (traffic ~320MB total, not 512MB).
// ---------------------------------------------------------------------------
__global__ void kpctx(const float* __restrict__ hs,
                      const float* __restrict__ attn,
                      float* __restrict__ pctx) {
    __shared__ float a_lds[SS / NSC];              // 128 attn weights
    const int b  = (BB - 1) - blockIdx.y;          // reverse batch walk
    const int sc = blockIdx.x;                     // 0..15
    const int t  = threadIdx.x;                    // 0..255

    if (t < SS / NSC)
        a_lds[t] = attn[(size_t)b * SS + sc * (SS / NSC) + t];
    __syncthreads();

    const float4* base =
        (const float4*)(hs + ((size_t)b * SS + (size_t)sc * (SS / NSC)) * HH) + t;
    float4 acc = {0.f, 0.f, 0.f, 0.f};
#pragma unroll 4
    for (int s = 0; s < SS / NSC; ++s) {
        float4 x = base[(size_t)s * (HH / 4)];     // 4KB contiguous per block step
        const float w = a_lds[s];
        acc.x += w * x.x; acc.y += w * x.y; acc.z += w * x.z; acc.w += w * x.w;
    }
    ((float4*)(pctx + ((size_t)b * NSC + sc) * HH))[t] = acc;
}

// ---------------------------------------------------------------------------
// Kernel 5: pre[b][0:1024]   = sum_sc pctx[b][sc][:]   (context)
//           pre[b][1024:2048] = hs[b, S-1, :]          (h_t)
// ---------------------------------------------------------------------------
__global__ void kreduce_pre(const float* __restrict__ hs,
                            const float* __restrict__ pctx,
                            float* __restrict__ pre) {
    const int tid = blockIdx.x * 256 + threadIdx.x; // 0..65535
    const int b = tid >> 11;
    const int k = tid & 2047;
    float val;
    if (k < HH) {
        float s = 0.f;
#pragma unroll
        for (int sc = 0; sc < NSC; ++sc)
            s += pctx[((size_t)b * NSC + sc) * HH + k];
        val = s;
    } else {
        val = hs[((size_t)b * SS + (SS - 1)) * HH + (k - HH)];
    }
    pre[tid] = val;
}

// ---------------------------------------------------------------------------
// Kernel 6: out = tanh(pre @ w_out). (32x2048)x(2048x128) -> 2x8 = 16 WMMA
// tiles, one block of 16 waves, one tile per wave, K = 2048.
// ---------------------------------------------------------------------------
__global__ void kout_wmma(const float* __restrict__ pre,
                          const float* __restrict__ w_out,
                          float* __restrict__ out) {
    const int lane = threadIdx.x & 31;
    const int wave = threadIdx.x >> 5;   // 0..15
    const int mt = wave >> 3;            // 0..1
    const int nt = wave & 7;             // 0..7

    const int m = mt * 16 + (lane & 15);
    const int n = nt * 16 + (lane & 15);
    const int koff = (lane >> 4) * 2;

    const float* arow = pre + (size_t)m * (2 * HH);
    v8f acc = {};
    for (int k = 0; k < 2 * HH; k += 4) {
        v2f a = *(const v2f*)(arow + k + koff);
        v2f b;
        b[0] = w_out[(size_t)(k + koff)     * UU + n];
        b[1] = w_out[(size_t)(k + koff + 1) * UU + n];
        acc = __builtin_amdgcn_wmma_f32_16x16x4_f32(
            false, a, false, b, (short)0, acc, false, false);
    }

    const int ncol  = nt * 16 + (lane & 15);
    const int mbase = mt * 16 + ((lane >> 4) ? 8 : 0);
#pragma unroll
    for (int i = 0; i < 8; ++i)
        out[(size_t)(mbase + i) * UU + ncol] = tanhf(acc[i]);
}

// ---------------------------------------------------------------------------
// Workspace layout (floats):
//   v    : 32*1024             = 32768
//   attn : 32*2048             = 65536   (scores, then softmaxed in place)
//   pctx : 32*16*1024          = 524288
//   pre  : 32*2048             = 65536
// total 688128 floats = ~2.75 MB
// ---------------------------------------------------------------------------
#define OFF_V    0
#define OFF_ATTN (OFF_V + BB * HH)
#define OFF_PCTX (OFF_ATTN + BB * SS)
#define OFF_PRE  (OFF_PCTX + BB * NSC * HH)

extern "C" void kernel_launch(void* const* d_in, const int* in_sizes, int n_in,
                              void* d_out, int out_size, void* d_ws, size_t ws_size,
                              hipStream_t stream) {
    const float* hs      = (const float*)d_in[0];  // (32, 2048, 1024) f32
    const float* w_score = (const float*)d_in[1];  // (1024, 1024) f32
    const float* w_out   = (const float*)d_in[2];  // (2048, 128) f32
    float* out = (float*)d_out;                    // (32, 128) f32
    float* ws  = (float*)d_ws;

    float* v    = ws + OFF_V;
    float* attn = ws + OFF_ATTN;
    float* pctx = ws + OFF_PCTX;
    float* pre  = ws + OFF_PRE;

    // 1) v = h_t @ w_score^T  (WMMA)
    kv_wmma<<<16, 256, 0, stream>>>(hs, w_score, v);
    // 2) score[b,s] = <hs[b,s,:], v[b,:]>   (streaming pass 1, ascending b)
    kscore<<<dim3(SS / 64, BB), 256, 0, stream>>>(hs, v, attn);
    // 3) softmax over s, in place
    ksoftmax<<<BB, 256, 0, stream>>>(attn);
    // 4) partial context  (streaming pass 2, descending b for L2 reuse)
    kpctx<<<dim3(NSC, BB), 256, 0, stream>>>(hs, attn, pctx);
    // 5) reduce partials + concat h_t -> pre
    kreduce_pre<<<(BB * SS) / 256, 256, 0, stream>>>(hs, pctx, pre);
    // 6) out = tanh(pre @ w_out)  (WMMA)
    kout_wmma<<<1, 512, 0, stream>>>(pre, w_out, out);
}